// loss_yolov1_36103495090636
// MI455X (gfx1250) — compile-verified
//
#include <hip/hip_runtime.h>
#include <stdint.h>

#define NCELLS     802816        // 4096 * 14 * 14
#define CPB        256           // cells (= threads) per block
#define NBLK       3136          // NCELLS / CPB
#define TILE_DW    7680          // 256 cells * 30 dwords
#define TILE_BYTES 30720
#define INV_S      (1.0f / 14.0f)

#if defined(__has_builtin)
#  if __has_builtin(__builtin_amdgcn_tensor_load_to_lds) && \
      __has_builtin(__builtin_amdgcn_s_wait_tensorcnt)
#    define YOLO_TDM 1
#  endif
#endif

typedef __attribute__((ext_vector_type(4))) unsigned int yolo_v4u;
typedef __attribute__((ext_vector_type(8))) int          yolo_v8i;
typedef __attribute__((ext_vector_type(4))) int          yolo_v4i;

#if defined(YOLO_TDM)
// 1-D contiguous DMA: copy TILE_BYTES from gaddr to LDS byte offset lds_byte_addr.
// D# group0: count=1, lds_addr, 57-bit global addr, type=2 ("image").
// D# group1: data_size=4B, tensor_dim0 = tile_dim0 = 7680 dwords, dim1 unused.
__device__ __forceinline__ void tdm_load_tile(unsigned long long gaddr,
                                              unsigned int lds_byte_addr) {
  yolo_v4u g0 = {1u,                                   // count=1 (user descriptor)
                 lds_byte_addr,                        // [63:32] lds_addr
                 (unsigned int)gaddr,                  // [95:64] global_addr lo
                 (unsigned int)(gaddr >> 32) | (2u << 30)}; // addr hi | type=2
  yolo_v8i g1 = {0x00020000,                // workgroup_mask=0, data_size=2 (4B)
                 (int)(TILE_DW << 16),      // tensor_dim0[15:0] at bits 63:48
                 0x00010000,                // tensor_dim0 hi=0, tensor_dim1=1
                 (int)(TILE_DW << 16),      // tensor_dim1 hi=0, tile_dim0=7680
                 0,                         // tile_dim1=0, tile_dim2=0 (1-D tile)
                 TILE_DW,                   // tensor_dim0_stride lo
                 0, 0};
  yolo_v4i gz = {0, 0, 0, 0};
#if __clang_major__ >= 23
  yolo_v8i gz8 = {0, 0, 0, 0, 0, 0, 0, 0};
  __builtin_amdgcn_tensor_load_to_lds(g0, g1, gz, gz, gz8, 0);
#else
  __builtin_amdgcn_tensor_load_to_lds(g0, g1, gz, gz, 0);
#endif
}
#endif

__global__ __launch_bounds__(CPB) void yolo_main(const float* __restrict__ pred,
                                                 const float* __restrict__ targ,
                                                 float* __restrict__ ws) {
  __shared__ float lds[2 * TILE_DW];   // [0,7680) pred tile, [7680,15360) targ tile
  __shared__ float red[8][5];
  const int t = threadIdx.x;
  const int tile = blockIdx.x;

  // Opaque never-taken store: compiler must assume lds[] may be written
  // (the TDM engine writes it behind the compiler's back).
  unsigned zr;
  asm volatile("s_mov_b32 %0, 0" : "=s"(zr));
  if (zr) { lds[t] = 0.0f; lds[TILE_DW + t] = 0.0f; }

#if defined(YOLO_TDM)
  if (t == 0) {
    unsigned int base = (unsigned int)(uintptr_t)&lds[0];
    tdm_load_tile((unsigned long long)(uintptr_t)pred +
                      (unsigned long long)tile * TILE_BYTES, base);
    tdm_load_tile((unsigned long long)(uintptr_t)targ +
                      (unsigned long long)tile * TILE_BYTES, base + TILE_BYTES);
  }
  __builtin_amdgcn_s_wait_tensorcnt(0);
  __syncthreads();
#else
  {
    const float2* gp = (const float2*)(pred + (size_t)tile * TILE_DW);
    const float2* gt = (const float2*)(targ + (size_t)tile * TILE_DW);
    float2* lp = (float2*)&lds[0];
    float2* lq = (float2*)&lds[TILE_DW];
#pragma unroll
    for (int i = 0; i < 15; ++i) {        // 15 * 256 = 3840 float2 per tensor
      int q = i * CPB + t;                // lane-contiguous -> coalesced B64
      lp[q] = gp[q];
      lq[q] = gt[q];
    }
  }
  __syncthreads();
#endif

  // ---- per-cell loss (one cell per thread, conflict-free LDS reads) ----
  const float* pc = &lds[t * 30];
  const float* tc = &lds[TILE_DW + t * 30];
  const int c = tile * CPB + t;
  const int rem = c % 196;                 // (i,j) within 14x14 grid
  const float offx = (float)(rem % 14) * INV_S;   // off = (j, i)
  const float offy = (float)(rem / 14) * INV_S;

  const float txv = tc[0], tyv = tc[1], twv = tc[2], thv = tc[3], tob = tc[4];
  const float m  = (tob == 1.0f) ? 1.0f : 0.0f;
  const float nm = (tob == 0.0f) ? 1.0f : 0.0f;

  const float tcx = txv * INV_S + offx, tcy = tyv * INV_S + offy;
  const float tx0 = tcx - 0.5f * twv, tx1 = tcx + 0.5f * twv;
  const float ty0 = tcy - 0.5f * thv, ty1 = tcy + 0.5f * thv;
  const float areat = twv * thv;

  auto iou_of = [&](float px, float py, float pw, float ph) -> float {
    float pcx = px * INV_S + offx, pcy = py * INV_S + offy;
    float x0 = pcx - 0.5f * pw, x1 = pcx + 0.5f * pw;
    float y0 = pcy - 0.5f * ph, y1 = pcy + 0.5f * ph;
    float iw = fmaxf(fminf(tx1, x1) - fmaxf(tx0, x0), 0.0f);
    float ih = fmaxf(fminf(ty1, y1) - fmaxf(ty0, y0), 0.0f);
    float inter = iw * ih;
    return inter / (areat + pw * ph - inter);
  };

  const float px0 = pc[0], py0 = pc[1], pw0 = pc[2], ph0 = pc[3], cf0 = pc[4];
  const float px1 = pc[5], py1 = pc[6], pw1 = pc[7], ph1 = pc[8], cf1 = pc[9];
  const float i0 = iou_of(px0, py0, pw0, ph0);
  const float i1 = iou_of(px1, py1, pw1, ph1);
  const bool k1 = (i1 > i0);               // argmax, first index wins ties
  const float bx = k1 ? px1 : px0, by = k1 ? py1 : py0;
  const float bw = k1 ? pw1 : pw0, bh = k1 ? ph1 : ph0;
  const float bc = k1 ? cf1 : cf0, biou = k1 ? i1 : i0;
  const float oc = k1 ? cf0 : cf1;         // non-best box confidence

  const float dx = txv - bx, dy = tyv - by;
  float sxy = m * (dx * dx + dy * dy);
  const float dw = sqrtf(twv) - sqrtf(bw), dh = sqrtf(thv) - sqrtf(bh);
  float swh = m * (dw * dw + dh * dh);
  const float dob = biou - bc;
  float sob = m * dob * dob;
  float sno = nm * (cf0 * cf0 + cf1 * cf1) + m * (oc * oc);
  float scl = 0.0f;
#pragma unroll
  for (int w = 10; w < 30; ++w) { float d = tc[w] - pc[w]; scl += d * d; }
  scl *= m;

  // ---- wave32 reduction, then block partials to d_ws (no global atomics) ----
#pragma unroll
  for (int o = 16; o > 0; o >>= 1) {
    sxy += __shfl_xor(sxy, o, 32);
    swh += __shfl_xor(swh, o, 32);
    sob += __shfl_xor(sob, o, 32);
    sno += __shfl_xor(sno, o, 32);
    scl += __shfl_xor(scl, o, 32);
  }
  const int wave = t >> 5, lane = t & 31;
  if (lane == 0) {
    red[wave][0] = sxy; red[wave][1] = swh; red[wave][2] = sob;
    red[wave][3] = sno; red[wave][4] = scl;
  }
  __syncthreads();
  if (t < 5) {
    float s = 0.0f;
#pragma unroll
    for (int w = 0; w < 8; ++w) s += red[w][t];
    ws[(size_t)tile * 8 + t] = s;
  }
}

__global__ __launch_bounds__(1024) void yolo_fin(const float* __restrict__ ws,
                                                 float* __restrict__ out) {
  __shared__ float red[1024];
  __shared__ float comp[5];
  const int t = threadIdx.x;
  for (int c = 0; c < 5; ++c) {
    float s = 0.0f;
    for (int i = t; i < NBLK; i += 1024) s += ws[(size_t)i * 8 + c];
    red[t] = s;
    __syncthreads();
    for (int k = 512; k > 0; k >>= 1) {
      if (t < k) red[t] += red[t + k];
      __syncthreads();
    }
    if (t == 0) comp[c] = red[0];
    __syncthreads();
  }
  if (t == 0) {
    const float invN = 1.0f / 4096.0f;
    const float lxy = comp[0] * 5.0f * invN;
    const float lwh = comp[1] * 5.0f * invN;
    const float lob = comp[2] * invN;
    const float lno = comp[3] * 0.5f * invN;
    const float lcl = comp[4] * invN;
    out[0] = lxy + lwh + lob + lno + lcl;
    out[1] = lxy; out[2] = lwh; out[3] = lob; out[4] = lno; out[5] = lcl;
  }
}

extern "C" void kernel_launch(void* const* d_in, const int* in_sizes, int n_in,
                              void* d_out, int out_size, void* d_ws, size_t ws_size,
                              hipStream_t stream) {
  (void)in_sizes; (void)n_in; (void)out_size; (void)ws_size;
  const float* pred = (const float*)d_in[0];
  const float* targ = (const float*)d_in[1];
  float* out = (float*)d_out;
  float* ws  = (float*)d_ws;   // needs NBLK*8*4 = 100,352 bytes
  hipLaunchKernelGGL(yolo_main, dim3(NBLK), dim3(CPB), 0, stream, pred, targ, ws);
  hipLaunchKernelGGL(yolo_fin, dim3(1), dim3(1024), 0, stream, ws, out);
}